// WorldNet_81887846465828
// MI455X (gfx1250) — compile-verified
//
#include <hip/hip_runtime.h>
#include <math.h>

typedef __attribute__((ext_vector_type(2))) float v2f;
typedef __attribute__((ext_vector_type(8))) float v8f;

#define STATE 115
#define ACT   39
#define INDIM 154
#define OUTC  116
#define BATCH 2048
#define TLEN  128

#define WAVES_PER_WG  2
#define ROWS_PER_WAVE 16

// 10 "layers": 9 MLP layers + reward head (115 -> 3).
// Kp = K padded to mult of 4, Np = N padded to mult of 16.
static constexpr int KDIM[10] = {154, 16, 32,  64, 128, 64, 32, 16,   8, 115};
static constexpr int NDIM[10] = { 16, 32, 64, 128,  64, 32, 16,  8, 115,   3};
static constexpr int KT_A[10] = { 39,  4,  8,  16,  32, 16,  8,  4,   2,  29};
// fragment region offsets (floats): frag size = KT*NT*64
static constexpr int FRAG_OFF[10] = {0, 2496, 3008, 5056, 13248, 21440, 23488, 24000, 24256, 25280};
static constexpr int BIAS_OFF[10] = {0, 16, 48, 112, 240, 304, 336, 352, 368, 496};
#define FRAG_TOTAL 27136
#define BIAS_TOTAL 512

#define X_FLOATS   (16*160)
#define PQ_FLOATS  (16*128)
#define R_FLOATS   (16*16)
#define WAVE_FLOATS (X_FLOATS + 2*PQ_FLOATS + R_FLOATS)              // 6912
#define LDS_FLOATS  (FRAG_TOTAL + BIAS_TOTAL + WAVES_PER_WG*WAVE_FLOATS) // 41472

// One GEMM layer on a 16-row batch tile through v_wmma_f32_16x16x4_f32.
// Split-K: even/odd kt feed independent accumulators to halve the dependent
// WMMA chain; N-tiles unrolled x2 for additional matrix-pipe ILP.
__device__ __forceinline__ void run_layer(const float* frag, const float* bias,
                                          const float* in, int inStride,
                                          float* outp, int outStride,
                                          int KT, int NT, bool clip, int lane)
{
    const int nl = lane & 15;      // row (A) / column (B,C,D) within tile
    const int hi = lane >> 4;      // K half-select for A, row half for C/D
    const float* aBase = in + nl * inStride + hi * 2;
#pragma unroll 2
    for (int nt = 0; nt < NT; ++nt) {
        const float bv = bias[nt * 16 + nl];
        v8f acc0 = {bv, bv, bv, bv, bv, bv, bv, bv};
        v8f acc1 = {0.f, 0.f, 0.f, 0.f, 0.f, 0.f, 0.f, 0.f};
        const float* fB = frag + nt * KT * 64 + lane * 2;
        int kt = 0;
        for (; kt + 1 < KT; kt += 2) {
            v2f a0; a0.x = aBase[kt * 4];     a0.y = aBase[kt * 4 + 1];
            v2f w0; w0.x = fB[kt * 64];       w0.y = fB[kt * 64 + 1];
            v2f a1; a1.x = aBase[kt * 4 + 4]; a1.y = aBase[kt * 4 + 5];
            v2f w1; w1.x = fB[kt * 64 + 64];  w1.y = fB[kt * 64 + 65];
            acc0 = __builtin_amdgcn_wmma_f32_16x16x4_f32(
                false, a0, false, w0, (short)0, acc0, false, false);
            acc1 = __builtin_amdgcn_wmma_f32_16x16x4_f32(
                false, a1, false, w1, (short)0, acc1, false, false);
        }
        if (kt < KT) {  // odd KT tail (layers 0 and 9)
            v2f a; a.x = aBase[kt * 4]; a.y = aBase[kt * 4 + 1];
            v2f w; w.x = fB[kt * 64];   w.y = fB[kt * 64 + 1];
            acc0 = __builtin_amdgcn_wmma_f32_16x16x4_f32(
                false, a, false, w, (short)0, acc0, false, false);
        }
#pragma unroll
        for (int r = 0; r < 8; ++r) {
            float v = acc0[r] + acc1[r];
            if (clip) v = fminf(fmaxf(v, 0.0f), 6.0f);
            outp[(r + hi * 8) * outStride + nt * 16 + nl] = v;
        }
    }
}

extern "C" __global__ void __launch_bounds__(32 * WAVES_PER_WG, 1)
worldnet_rollout(const float* __restrict__ u,
                 const float* __restrict__ W0, const float* __restrict__ b0,
                 const float* __restrict__ W1, const float* __restrict__ b1,
                 const float* __restrict__ W2, const float* __restrict__ b2,
                 const float* __restrict__ W3, const float* __restrict__ b3,
                 const float* __restrict__ W4, const float* __restrict__ b4,
                 const float* __restrict__ W5, const float* __restrict__ b5,
                 const float* __restrict__ W6, const float* __restrict__ b6,
                 const float* __restrict__ W7, const float* __restrict__ b7,
                 const float* __restrict__ W8, const float* __restrict__ b8,
                 const float* __restrict__ Wr, const float* __restrict__ br,
                 float* __restrict__ out)
{
    extern __shared__ float lds[];
    const int tid = threadIdx.x;
    const int nth = blockDim.x;

    // ---- stage weights into fragment-swizzled LDS (once, reused 128 steps) ----
    const float* Wp[10] = {W0, W1, W2, W3, W4, W5, W6, W7, W8, Wr};
    const float* bp[10] = {b0, b1, b2, b3, b4, b5, b6, b7, b8, br};
    for (int i = tid; i < FRAG_TOTAL + BIAS_TOTAL; i += nth) lds[i] = 0.0f;
    __syncthreads();
    for (int l = 0; l < 10; ++l) {
        const int Kd = KDIM[l], Nd = NDIM[l], KT = KT_A[l];
        const float* W = Wp[l];
        float* fb = lds + FRAG_OFF[l];
        for (int i = tid; i < Kd * Nd; i += nth) {
            const int k = i / Nd, n = i - k * Nd;
            const int kt = k >> 2, kl = k & 3, ntile = n >> 4, nl = n & 15;
            const int ln = ((kl >> 1) << 4) | nl;   // B layout: K rows 0,1 lanes 0-15; 2,3 lanes 16-31
            fb[((ntile * KT + kt) * 32 + ln) * 2 + (kl & 1)] = W[i];
        }
        float* bb = lds + FRAG_TOTAL + BIAS_OFF[l];
        for (int n = tid; n < Nd; n += nth) bb[n] = bp[l][n];
    }
    __syncthreads();

    // ---- per-wave state ----
    const int wave = tid >> 5;
    const int lane = tid & 31;
    const int row0 = (blockIdx.x * WAVES_PER_WG + wave) * ROWS_PER_WAVE;

    const float* FRG = lds;
    const float* BIA = lds + FRAG_TOTAL;
    float* Xb = lds + FRAG_TOTAL + BIAS_TOTAL + wave * WAVE_FLOATS; // 16 x 160 (state+act, zero-padded)
    float* Pb = Xb + X_FLOATS;                                      // 16 x 128 ping
    float* Qb = Pb + PQ_FLOATS;                                     // 16 x 128 pong
    float* Rb = Qb + PQ_FLOATS;                                     // 16 x 16 reward tile

    // zero X (covers pad cols 154..159), then load x0 = u[:,0,:115]
    for (int i = lane; i < X_FLOATS; i += 32) Xb[i] = 0.0f;
    for (int m = 0; m < 16; ++m)
        for (int c = lane; c < STATE; c += 32)
            Xb[m * 160 + c] = u[(size_t)(row0 + m) * TLEN * INDIM + c];

    for (int t = 0; t < TLEN; ++t) {
        // load actions for this step; store x_t (pre-update state) to output
        for (int m = 0; m < 16; ++m) {
            const float* urow = u + ((size_t)(row0 + m) * TLEN + t) * INDIM;
            float* orow = out + ((size_t)(row0 + m) * TLEN + t) * OUTC;
            for (int c = lane; c < ACT; c += 32) Xb[m * 160 + STATE + c] = urow[STATE + c];
            for (int c = lane; c < STATE; c += 32) orow[c] = Xb[m * 160 + c];
        }

        // 9-layer MLP, all through fp32 WMMA
        run_layer(FRG + FRAG_OFF[0], BIA + BIAS_OFF[0], Xb, 160, Pb, 128, 39, 1, true,  lane);
        run_layer(FRG + FRAG_OFF[1], BIA + BIAS_OFF[1], Pb, 128, Qb, 128,  4, 2, true,  lane);
        run_layer(FRG + FRAG_OFF[2], BIA + BIAS_OFF[2], Qb, 128, Pb, 128,  8, 4, true,  lane);
        run_layer(FRG + FRAG_OFF[3], BIA + BIAS_OFF[3], Pb, 128, Qb, 128, 16, 8, true,  lane);
        run_layer(FRG + FRAG_OFF[4], BIA + BIAS_OFF[4], Qb, 128, Pb, 128, 32, 4, true,  lane);
        run_layer(FRG + FRAG_OFF[5], BIA + BIAS_OFF[5], Pb, 128, Qb, 128, 16, 2, true,  lane);
        run_layer(FRG + FRAG_OFF[6], BIA + BIAS_OFF[6], Qb, 128, Pb, 128,  8, 1, true,  lane);
        run_layer(FRG + FRAG_OFF[7], BIA + BIAS_OFF[7], Pb, 128, Qb, 128,  4, 1, true,  lane);
        run_layer(FRG + FRAG_OFF[8], BIA + BIAS_OFF[8], Qb, 128, Pb, 128,  2, 8, false, lane);

        // residual: x_{t+1} = x_t + mlp_out ; also stage reward input (116 cols, last = 0)
        for (int m = 0; m < 16; ++m)
            for (int c = lane; c < STATE; c += 32) {
                const float xn = Xb[m * 160 + c] + Pb[m * 128 + c];
                Xb[m * 160 + c] = xn;
                Qb[m * 128 + c] = xn;
            }
        if (lane < 16) Qb[lane * 128 + STATE] = 0.0f;

        // reward head: (16 x 116) @ (116 x 3) via WMMA (K padded 115->116, N padded 3->16)
        run_layer(FRG + FRAG_OFF[9], BIA + BIAS_OFF[9], Qb, 128, Rb, 16, 29, 1, false, lane);

        if (lane < 16) {
            const int m = lane;
            const float r0 = Rb[m * 16 + 0], r1 = Rb[m * 16 + 1], r2 = Rb[m * 16 + 2];
            out[((size_t)(row0 + m) * TLEN + t) * OUTC + STATE] =
                -sqrtf(r0 * r0 + r1 * r1 + r2 * r2);
        }
    }
}

extern "C" void kernel_launch(void* const* d_in, const int* in_sizes, int n_in,
                              void* d_out, int out_size, void* d_ws, size_t ws_size,
                              hipStream_t stream) {
    const float* u = (const float*)d_in[0];
    const float* W[9];
    const float* b[9];
    for (int i = 0; i < 9; ++i) {
        W[i] = (const float*)d_in[1 + 2 * i];
        b[i] = (const float*)d_in[2 + 2 * i];
    }
    const float* Wr = (const float*)d_in[19];
    const float* br = (const float*)d_in[20];
    float* outp = (float*)d_out;

    const size_t shmem = (size_t)LDS_FLOATS * sizeof(float); // ~162 KB < 320 KB/WG
    hipFuncSetAttribute(reinterpret_cast<const void*>(worldnet_rollout),
                        hipFuncAttributeMaxDynamicSharedMemorySize, (int)shmem);

    const int grid = BATCH / (ROWS_PER_WAVE * WAVES_PER_WG); // 64 blocks
    worldnet_rollout<<<dim3(grid), dim3(32 * WAVES_PER_WG), shmem, stream>>>(
        u,
        W[0], b[0], W[1], b[1], W[2], b[2], W[3], b[3], W[4], b[4],
        W[5], b[5], W[6], b[6], W[7], b[7], W[8], b[8],
        Wr, br, outp);
}